// PhysicsLoss_43619687858928
// MI455X (gfx1250) — compile-verified
//
#include <hip/hip_runtime.h>

typedef float v2f __attribute__((ext_vector_type(2)));
typedef float v8f __attribute__((ext_vector_type(8)));

#define S   96
#define S2  (96*96)
#define S3  (96*96*96)
#define TDIM 8
#define TB  7             // valid timesteps t=1..7
#define NBATCH 2
#define YCHUNK 48         // rows marched per block (2 chunks per column)
#define INV_DT 100.0f     // 1/0.01
#define INVH   10.0f      // 1/h, h=0.1
#define INV2H  5.0f       // 1/(2h)
#define DEFF   0.001f
#define NPOINTS 12386304.0f   // 2*7*96^3

__device__ __forceinline__ int   clampi(int i) { return i < 0 ? 0 : (i > S - 1 ? S - 1 : i); }
// jnp.gradient weight: 1/h at the two edges, 1/(2h) in the interior
__device__ __forceinline__ float gw(int i)     { return (i == 0 || i == S - 1) ? INVH : INV2H; }

// Each thread marches a y-column at fixed (x,z); blockDim=(96,4) = 12 wave32's.
__global__ __launch_bounds__(384)
void physics_loss_partial(const float* __restrict__ c_pred,
                          const float* __restrict__ u_future,
                          const float* __restrict__ mask,
                          float* __restrict__ partials) {
    const int x = threadIdx.x;                       // 0..95, lane-contiguous
    int id = blockIdx.x;
    const int yc = id % (S / YCHUNK); id /= (S / YCHUNK);
    const int zg = id % (S / 4);      id /= (S / 4);
    const int tt = id % TB;           id /= TB;
    const int b  = id;
    const int t  = tt + 1;
    const int z  = zg * 4 + threadIdx.y;             // wave-uniform
    const int y0 = yc * YCHUNK;

    const int c_t_base   = (b*TDIM + t)     * S3 + z * S2;   // + y*S + x
    const int c_tm1_base = (b*TDIM + t - 1) * S3 + z * S2;
    const int ux_base    = ((b*TDIM + t)*3    ) * S3 + z * S2;
    const int uy_base    = ((b*TDIM + t)*3 + 1) * S3 + z * S2;
    const int uz_base    = ((b*TDIM + t)*3 + 2) * S3 + z * S2;
    const int m_base     = b * S3 + z * S2;

    // ---- x axis: lane-variant but loop-invariant clamp deltas / weights / masks ----
    const int   xp = clampi(x+1), xm = clampi(x-1);
    const int   dxp  = xp - x,          dxm  = xm - x;
    const int   dxp2 = clampi(x+2) - x, dxm2 = clampi(x-2) - x;
    const float wx = gw(x), wxp = gw(xp), wxm = gw(xm);
    const bool  xhi = (x == S-1), xlo = (x == 0);

    // ---- z axis: wave-uniform, loop-invariant ----
    const int   zp = clampi(z+1), zm = clampi(z-1);
    const int   dzp  = (zp - z) * S2,          dzm  = (zm - z) * S2;
    const int   dzp2 = (clampi(z+2) - z) * S2, dzm2 = (clampi(z-2) - z) * S2;
    const float wz = gw(z), wzp = gw(zp), wzm = gw(zm);
    const bool  zhi = (z == S-1), zlo = (z == 0);

    // ---- y axis: sliding register windows (clamped rows) ----
#define CROW(j)  c_pred[c_t_base + clampi(j) * S + x]
#define UYROW(j) u_future[uy_base + clampi(j) * S + x]
    float w_m2 = CROW(y0-2), w_m1 = CROW(y0-1), w_0 = CROW(y0);
    float w_p1 = CROW(y0+1), w_p2 = CROW(y0+2);
    float uym = UYROW(y0-1), uy0 = UYROW(y0), uyp = UYROW(y0+1);

    float acc = 0.0f;

    for (int y = y0; y < y0 + YCHUNK; ++y) {
        const int row = y * S + x;
        const int ic  = c_t_base + row;

        const float ctm1 = c_pred[c_tm1_base + row];
        const float m    = mask[m_base + row];

        // x-axis samples (same cacheline neighborhoods -> L0 hits)
        const float cxp1 = c_pred[ic + dxp],  cxm1 = c_pred[ic + dxm];
        const float cxp2 = c_pred[ic + dxp2], cxm2 = c_pred[ic + dxm2];
        const float uxp  = u_future[ux_base + row + dxp];
        const float uxm  = u_future[ux_base + row + dxm];
        // z-axis samples
        const float czp1 = c_pred[ic + dzp],  czm1 = c_pred[ic + dzm];
        const float czp2 = c_pred[ic + dzp2], czm2 = c_pred[ic + dzm2];
        const float uzp  = u_future[uz_base + row + dzp];
        const float uzm  = u_future[uz_base + row + dzm];

        const float wy  = gw(y);
        const float wyp = gw(clampi(y+1)), wym = gw(clampi(y-1));
        const bool  yhi = (y == S-1), ylo = (y == 0);

        const float dcdt = (w_0 - ctm1) * INV_DT;

        // flux divergence d/dax(u_ax * c) with jnp.gradient edge handling
        const float divx = (uxp * cxp1 - uxm * cxm1) * wx;
        const float divy = (uyp * w_p1 - uym * w_m1) * wy;
        const float divz = (uzp * czp1 - uzm * czm1) * wz;

        // gradient(gradient(c)): inner clamp(ip-1)/clamp(im+1) == center except
        // at the outermost coordinate, where it collapses to the +-1 sample.
        const float lapx = ((cxp2 - (xhi ? cxm1 : w_0)) * wxp
                          - ((xlo ? cxp1 : w_0) - cxm2) * wxm) * wx;
        const float lapy = ((w_p2 - (yhi ? w_m1 : w_0)) * wyp
                          - ((ylo ? w_p1 : w_0) - w_m2) * wym) * wy;
        const float lapz = ((czp2 - (zhi ? czm1 : w_0)) * wzp
                          - ((zlo ? czp1 : w_0) - czm2) * wzm) * wz;

        const float r = (dcdt + (divx + divy + divz) - DEFF * (lapx + lapy + lapz)) * m;
        acc = fmaf(r, r, acc);

        // advance sliding windows (1 new c row + 1 new uy row per step)
        w_m2 = w_m1; w_m1 = w_0; w_0 = w_p1; w_p1 = w_p2;
        w_p2 = CROW(y + 3);
        uym = uy0; uy0 = uyp;
        uyp = UYROW(y + 2);

        // speculative prefetch a few rows ahead (clamped address, always valid)
        __builtin_prefetch(&CROW(y + 8), 0, 3);
    }
#undef CROW
#undef UYROW

    // ---- wave32 reduction using V_WMMA_F32_16X16X4_F32 ----
    // A = ones(16x4); 32 per-lane partials fill B's 64 slots (2nd slot zero).
    // D[0,n] = sum_k B[k,n]; summing a 16-lane group of D row 0 yields the exact
    // 32-lane wave total independent of the lane->(k,n) bijection.
    float wsum;
#if __has_builtin(__builtin_amdgcn_wmma_f32_16x16x4_f32)
    {
        v2f a;  a[0] = 1.0f; a[1] = 1.0f;
        v2f bm; bm[0] = acc; bm[1] = 0.0f;
        v8f cz = {};
        v8f d = __builtin_amdgcn_wmma_f32_16x16x4_f32(false, a, false, bm,
                                                      (short)0, cz, false, false);
        float v = d[0];
        v += __shfl_xor(v, 1, 32);
        v += __shfl_xor(v, 2, 32);
        v += __shfl_xor(v, 4, 32);
        v += __shfl_xor(v, 8, 32);
        wsum = v;
    }
#else
    {
        float v = acc;
        for (int off = 16; off > 0; off >>= 1) v += __shfl_xor(v, off, 32);
        wsum = v;
    }
#endif

    __shared__ float smem[12];
    const int tid  = threadIdx.y * S + threadIdx.x;
    const int wave = tid >> 5;
    if ((tid & 31) == 0) smem[wave] = wsum;
    __syncthreads();
    if (tid == 0) {
        float s = 0.0f;
        for (int w = 0; w < 12; ++w) s += smem[w];
        partials[blockIdx.x] = s;             // fixed-order sum -> deterministic
    }
}

// Deterministic final reduction of the per-block partials -> scalar mean.
__global__ __launch_bounds__(128)
void physics_loss_finalize(const float* __restrict__ partials, float* __restrict__ out, int n) {
    __shared__ float smem[128];
    float s = 0.0f;
    for (int i = threadIdx.x; i < n; i += 128) s += partials[i];  // fixed per-thread order
    smem[threadIdx.x] = s;
    __syncthreads();
    if (threadIdx.x == 0) {
        float tot = 0.0f;
        for (int i = 0; i < 128; ++i) tot += smem[i];
        out[0] = tot * (1.0f / NPOINTS);
    }
}

extern "C" void kernel_launch(void* const* d_in, const int* in_sizes, int n_in,
                              void* d_out, int out_size, void* d_ws, size_t ws_size,
                              hipStream_t stream) {
    (void)in_sizes; (void)n_in; (void)out_size; (void)ws_size;
    const float* c_pred   = (const float*)d_in[0];
    const float* u_future = (const float*)d_in[1];
    const float* mask     = (const float*)d_in[2];
    float* partials = (float*)d_ws;           // 672 floats = 2.6 KB scratch
    float* out      = (float*)d_out;

    const int nblocks = NBATCH * TB * (S / 4) * (S / YCHUNK);   // 672
    dim3 block(96, 4, 1);                     // 384 threads = 12 wave32's
    physics_loss_partial<<<nblocks, block, 0, stream>>>(c_pred, u_future, mask, partials);
    physics_loss_finalize<<<1, 128, 0, stream>>>(partials, out, nblocks);
}